// EventSupervisedLoss_47528108097572
// MI455X (gfx1250) — compile-verified
//
#include <hip/hip_runtime.h>
#include <hip/hip_bf16.h>

// ---------------- problem constants (match reference) ----------------
static constexpr int Bc = 1, Sc = 4, Hc = 224, Wc = 224;
static constexpr int HW   = Hc * Wc;        // 50176
static constexpr int NFR  = Bc * Sc;        // 4 frames
static constexpr int NS   = 4096;           // NUM_SAMPLES
static constexpr int EQCAP = 8192;          // tie buffer cap per frame
static constexpr int DN_BLK = 256;
static constexpr int DN_GRID = (NFR * HW + DN_BLK - 1) / DN_BLK; // 784
#define EPSF 1e-6f

typedef float v2f __attribute__((ext_vector_type(2)));
typedef float v8f __attribute__((ext_vector_type(8)));

// ---------------- small helpers ----------------
__device__ __forceinline__ bool finitef(float x) { return __builtin_isfinite(x); }

// raw hardware v_sqrt_f32 (no IEEE refinement fixup) for the hot Chamfer loop;
// inputs are clamped to [1e-12, ~1e9] so the denorm/overflow fixup is dead weight.
__device__ __forceinline__ float fast_sqrtf(float x) { return __builtin_amdgcn_sqrtf(x); }

__device__ __forceinline__ int block_reduce_int(int v) {
  __shared__ int sh[256];
  int t = threadIdx.x;
  sh[t] = v; __syncthreads();
  for (int s = 128; s > 0; s >>= 1) { if (t < s) sh[t] += sh[t + s]; __syncthreads(); }
  int r = sh[0]; __syncthreads();
  return r;
}

__device__ __forceinline__ float block_reduce_float(float v) {
  __shared__ float sh[256];
  int t = threadIdx.x;
  sh[t] = v; __syncthreads();
  for (int s = 128; s > 0; s >>= 1) { if (t < s) sh[t] += sh[t + s]; __syncthreads(); }
  float r = sh[0]; __syncthreads();
  return r;
}

__device__ __forceinline__ bool base_mask(float g) {
  return (g > 0.0f) && finitef(g) && (g > 1e-6f);
}

// masked ratio for median-scale (mask additionally requires finite dp > EPS)
__device__ __forceinline__ bool ratio_ok(const float* dpf, const float* dgf, int i, unsigned* u) {
  float g = dgf[i], pr = dpf[i];
  bool m = base_mask(g) && finitef(pr) && (pr > EPSF);
  if (m) *u = __float_as_uint(pr / fmaxf(g, EPSF));
  return m;
}

// top-k score bit pattern: score = mask*2 + jitter  (>= 0 => uint order == float order)
__device__ __forceinline__ unsigned score_bits(const float* dgf, int i) {
  float g = dgf[i];
  bool m = base_mask(g);
  float jit = (float)(((unsigned)i * 2654435761u) & 0xFFFFFu) * (1.0f / 1048576.0f);
  float s = (m ? 2.0f : 0.0f) + jit;
  return __float_as_uint(s);
}

// ---------------- init ----------------
__global__ void init_kernel(float* acc, unsigned* mingp, int* sel) {
  int t = blockIdx.x * blockDim.x + threadIdx.x;
  if (t < 16) acc[t] = 0.0f;
  if (t < NFR * NS) {
    mingp[t] = __float_as_uint(1e9f);
    sel[t] = 0;
  }
}

// ---------------- pose loss + relative GT transforms ----------------
__device__ void inv4(const float* A, float* out) {
  float M[4][8];
  for (int r = 0; r < 4; ++r)
    for (int c = 0; c < 4; ++c) { M[r][c] = A[r * 4 + c]; M[r][4 + c] = (r == c) ? 1.0f : 0.0f; }
  for (int col = 0; col < 4; ++col) {
    int piv = col; float best = fabsf(M[col][col]);
    for (int r = col + 1; r < 4; ++r) { float a = fabsf(M[r][col]); if (a > best) { best = a; piv = r; } }
    if (piv != col)
      for (int c = 0; c < 8; ++c) { float tmp = M[col][c]; M[col][c] = M[piv][c]; M[piv][c] = tmp; }
    float ip = 1.0f / M[col][col];
    for (int c = 0; c < 8; ++c) M[col][c] *= ip;
    for (int r = 0; r < 4; ++r) if (r != col) {
      float fk = M[r][col];
      for (int c = 0; c < 8; ++c) M[r][c] -= fk * M[col][c];
    }
  }
  for (int r = 0; r < 4; ++r)
    for (int c = 0; c < 4; ++c) out[r * 4 + c] = M[r][4 + c];
}

__device__ void mul4(const float* A, const float* B, float* C) {
  for (int r = 0; r < 4; ++r)
    for (int c = 0; c < 4; ++c) {
      float s = 0.0f;
      for (int k = 0; k < 4; ++k) s += A[r * 4 + k] * B[k * 4 + c];
      C[r * 4 + c] = s;
    }
}

__global__ void pose_kernel(const float* __restrict__ pp, const float* __restrict__ pg,
                            float* __restrict__ acc, float* __restrict__ relgt) {
  if (threadIdx.x != 0 || blockIdx.x != 0) return;
  float i0p[16], i0g[16];
  inv4(pp, i0p);
  inv4(pg, i0g);
  float sum = 0.0f;
  for (int s = 0; s < Sc; ++s) {
    float rp[16], rg[16];
    mul4(i0p, pp + s * 16, rp);
    mul4(i0g, pg + s * 16, rg);
    for (int e = 0; e < 16; ++e) sum += fabsf(rp[e] - rg[e]);
    for (int r = 0; r < 3; ++r)
      for (int c = 0; c < 4; ++c) relgt[s * 12 + r * 4 + c] = rg[r * 4 + c];
  }
  acc[0] = sum / (float)(Sc * 16); // mean over (1,4,4,4)
}

// ---------------- per-frame median ratio -> scale ----------------
__global__ void median_scale_kernel(const float* __restrict__ dp, const float* __restrict__ dgt,
                                    float* __restrict__ scale) {
  int f = blockIdx.x, t = threadIdx.x;
  const float* dpf = dp + (size_t)f * HW;
  const float* dgf = dgt + (size_t)f * HW;
  int c = 0;
  for (int i = t; i < HW; i += 256) { unsigned u; c += ratio_ok(dpf, dgf, i, &u) ? 1 : 0; }
  int cnt = block_reduce_int(c);
  float sc = 1.0f;
  if (cnt > 0) {
    int ks[2]; ks[0] = (cnt - 1) / 2; ks[1] = cnt / 2;
    unsigned bits[2];
    for (int q = 0; q < 2; ++q) {
      if (q == 1 && ks[1] == ks[0]) { bits[1] = bits[0]; break; }
      unsigned lo = 0u, hi = 0xFFFFFFFFu;
      for (int it = 0; it < 33 && lo < hi; ++it) {
        unsigned mid = lo + ((hi - lo) >> 1);
        int cl = 0;
        for (int i = t; i < HW; i += 256) {
          unsigned u;
          if (ratio_ok(dpf, dgf, i, &u) && u <= mid) cl++;
        }
        cl = block_reduce_int(cl);
        if (cl >= ks[q] + 1) hi = mid; else lo = mid + 1;
      }
      bits[q] = hi;
    }
    float med = 0.5f * (__uint_as_float(bits[0]) + __uint_as_float(bits[1]));
    if (cnt >= 16 && finitef(med)) sc = fminf(fmaxf(med, 0.001f), 1000.0f);
  }
  if (t == 0) scale[f] = sc;
}

// ---------------- depth L1 + normal cosine (per pixel, block partials) ----------------
__device__ __forceinline__ void cam_pt(float d, float xx, float yy,
                                       float fx, float fy, float cx, float cy, float o[3]) {
  o[0] = (xx - cx) / fx * d;
  o[1] = (yy - cy) / fy * d;
  o[2] = d;
}

__device__ __forceinline__ void normal_at(const float* __restrict__ D, float smul, int x, int y,
                                          float fx, float fy, float cx, float cy, float n[3]) {
  float pr[3], pl[3], pd[3], pu[3];
  cam_pt(D[y * Wc + (x + 1)] * smul, (float)(x + 1), (float)y, fx, fy, cx, cy, pr);
  cam_pt(D[y * Wc + (x - 1)] * smul, (float)(x - 1), (float)y, fx, fy, cx, cy, pl);
  cam_pt(D[(y + 1) * Wc + x] * smul, (float)x, (float)(y + 1), fx, fy, cx, cy, pd);
  cam_pt(D[(y - 1) * Wc + x] * smul, (float)x, (float)(y - 1), fx, fy, cx, cy, pu);
  float fxd[3] = { pr[0] - pl[0], pr[1] - pl[1], pr[2] - pl[2] };
  float fyd[3] = { pd[0] - pu[0], pd[1] - pu[1], pd[2] - pu[2] };
  n[0] = fyd[1] * fxd[2] - fyd[2] * fxd[1];
  n[1] = fyd[2] * fxd[0] - fyd[0] * fxd[2];
  n[2] = fyd[0] * fxd[1] - fyd[1] * fxd[0];
  float nn = sqrtf(n[0] * n[0] + n[1] * n[1] + n[2] * n[2]);
  float iv = 1.0f / fmaxf(nn, EPSF);
  n[0] *= iv; n[1] *= iv; n[2] *= iv;
  nn = sqrtf(n[0] * n[0] + n[1] * n[1] + n[2] * n[2]); // second normalize (masked_cosine)
  iv = 1.0f / fmaxf(nn, EPSF);
  n[0] *= iv; n[1] *= iv; n[2] *= iv;
}

__global__ void depth_normal_kernel(const float* __restrict__ dp, const float* __restrict__ dgt,
                                    const float* __restrict__ intr, const float* __restrict__ scale,
                                    float* __restrict__ part) {
  int gid = blockIdx.x * blockDim.x + threadIdx.x;
  float dl = 0.0f, nl = 0.0f, wv = 0.0f;
  if (gid < NFR * HW) {
    int f = gid / HW, pix = gid % HW;
    int y = pix / Wc, x = pix % Wc;
    const float* I = intr + f * 9;
    float fx = fmaxf(I[0], EPSF), fy = fmaxf(I[4], EPSF), cx = I[2], cy = I[5];
    float g = dgt[gid];
    bool m = base_mask(g);
    wv = m ? 1.0f : 0.0f;
    float s = scale[f];
    dl = fabsf(dp[gid] - g * s) * wv;
    if (x >= 1 && x < Wc - 1 && y >= 1 && y < Hc - 1) {
      float np_[3], nt[3];
      normal_at(dp + (size_t)f * HW, 1.0f, x, y, fx, fy, cx, cy, np_);
      normal_at(dgt + (size_t)f * HW, s, x, y, fx, fy, cx, cy, nt);
      float cosv = np_[0] * nt[0] + np_[1] * nt[1] + np_[2] * nt[2];
      cosv = fminf(fmaxf(cosv, -1.0f), 1.0f);
      nl = (1.0f - cosv) * wv;
    } else {
      nl = wv; // border normals are zero vectors -> cos = 0 -> (1-0)*w
    }
  }
  dl = block_reduce_float(dl);
  nl = block_reduce_float(nl);
  wv = block_reduce_float(wv);
  if (threadIdx.x == 0) {
    part[blockIdx.x * 4 + 0] = dl;
    part[blockIdx.x * 4 + 1] = nl;
    part[blockIdx.x * 4 + 2] = wv;
  }
}

// ---------------- top-NS sample selection per frame ----------------
__global__ void select_kernel(const float* __restrict__ dgt, int* __restrict__ sel,
                              int* __restrict__ eqbuf) {
  int f = blockIdx.x, t = threadIdx.x;
  const float* dgf = dgt + (size_t)f * HW;
  const int k = HW - NS; // threshold = k-th smallest (0-based) => smallest of the top NS
  unsigned lo = 0u, hi = 0xFFFFFFFFu;
  for (int it = 0; it < 33 && lo < hi; ++it) {
    unsigned mid = lo + ((hi - lo) >> 1);
    int cl = 0;
    for (int i = t; i < HW; i += 256) cl += (score_bits(dgf, i) <= mid) ? 1 : 0;
    cl = block_reduce_int(cl);
    if (cl >= k + 1) hi = mid; else lo = mid + 1;
  }
  unsigned T = hi;
  int cg = 0;
  for (int i = t; i < HW; i += 256) cg += (score_bits(dgf, i) > T) ? 1 : 0;
  int G = block_reduce_int(cg); // strictly greater than threshold (<= NS-1)

  __shared__ int s_sel, s_eq;
  if (t == 0) { s_sel = 0; s_eq = 0; }
  __syncthreads();
  for (int i = t; i < HW; i += 256) {
    unsigned u = score_bits(dgf, i);
    if (u > T) {
      int p = atomicAdd(&s_sel, 1);
      if (p < NS) sel[f * NS + p] = i;
    } else if (u == T) {
      int p = atomicAdd(&s_eq, 1);
      if (p < EQCAP) eqbuf[f * EQCAP + p] = i;
    }
  }
  __syncthreads();
  int Q = s_eq < EQCAP ? s_eq : EQCAP;
  int E = NS - G;
  if (E < 0) E = 0;
  if (E > Q) E = Q;
  // tie-break: smallest indices win (matches stable top_k); O(Q^2) with tiny Q
  for (int j = t; j < Q; j += 256) {
    int me = eqbuf[f * EQCAP + j];
    int rank = 0;
    for (int l = 0; l < Q; ++l) rank += (eqbuf[f * EQCAP + l] < me) ? 1 : 0;
    if (rank < E) {
      int p = atomicAdd(&s_sel, 1);
      if (p < NS) sel[f * NS + p] = me;
    }
  }
  __syncthreads();
  int total = s_sel < NS ? s_sel : NS;
  for (int p = total + t; p < NS; p += 256) sel[f * NS + p] = 0; // pathological underfill pad
}

// ---------------- gather sampled pred/gt world points ----------------
__global__ void gather_kernel(const float* __restrict__ dgt, const float* __restrict__ ppred,
                              const float* __restrict__ intr, const float* __restrict__ scale,
                              const float* __restrict__ relgt, const int* __restrict__ sel,
                              float4* __restrict__ Ppts, float4* __restrict__ Gpts,
                              float* __restrict__ Msel) {
  int gid = blockIdx.x * blockDim.x + threadIdx.x;
  if (gid >= NFR * NS) return;
  int f = gid / NS;
  int pix = sel[gid];
  const float* I = intr + f * 9;
  float fx = fmaxf(I[0], EPSF), fy = fmaxf(I[4], EPSF), cx = I[2], cy = I[5];
  float g = dgt[(size_t)f * HW + pix];
  bool m = base_mask(g);
  float d = g * scale[f];
  int px = pix % Wc, py = pix / Wc;
  float X = ((float)px - cx) / fx * d;
  float Y = ((float)py - cy) / fy * d;
  const float* M = relgt + f * 12;
  float wx = M[0] * X + M[1] * Y + M[2] * d + M[3];
  float wy = M[4] * X + M[5] * Y + M[6] * d + M[7];
  float wz = M[8] * X + M[9] * Y + M[10] * d + M[11];
  Gpts[gid] = make_float4(wx, wy, wz, wx * wx + wy * wy + wz * wz);
  const float* pp = ppred + ((size_t)f * HW + pix) * 3;
  float ax = pp[0], ay = pp[1], az = pp[2];
  Ppts[gid] = make_float4(ax, ay, az, ax * ax + ay * ay + az * az);
  Msel[gid] = m ? 1.0f : 0.0f;
}

// ---------------- Chamfer with WMMA f32 16x16x4 ----------------
// Each wave owns a fixed 16-row A tile (P samples), loops over all 4096 G columns.
// d2(i,j) = |p_i|^2 + |g_j|^2 - 2 * (P G^T)(i,j); P G^T via V_WMMA_F32_16X16X4_F32 (K=3, pad 0).
// A 16x4 f32 layout: lane L holds row M=L%16; K={0,1} for L<16, K={2,3} for L>=16 (K=3 -> 0).
// B 4x16 f32 layout: lane L holds col N=L%16; VGPR0: K = L<16?0:2, VGPR1: K = L<16?1:3.
// C/D 16x16 f32 layout: VGPR e -> row M = e + (L>=16 ? 8 : 0), col N = L%16.
__global__ void __launch_bounds__(256)
chamfer_kernel(const float4* __restrict__ Ppts, const float4* __restrict__ Gpts,
               const float* __restrict__ Msel, float* __restrict__ minpg,
               unsigned* __restrict__ mingp) {
  const int f = blockIdx.x / (NS / 128);
  const int stripe = blockIdx.x % (NS / 128);
  const int wave = threadIdx.x >> 5;
  const int lane = threadIdx.x & 31;
  const int laneN = lane & 15;
  const bool hiK = lane >= 16;
  const int rowbase = stripe * 128 + wave * 16;

  // A fragment (held for the whole column loop)
  float4 Prow = Ppts[f * NS + rowbase + laneN];
  v2f a;
  a.x = hiK ? Prow.z : Prow.x;
  a.y = hiK ? 0.0f : Prow.y;

  float pn2[8], rm[8], rowmin[8];
#pragma unroll
  for (int e = 0; e < 8; ++e) {
    int r = rowbase + e + (hiK ? 8 : 0);
    float4 pv = Ppts[f * NS + r];
    pn2[e] = pv.w;
    rm[e] = Msel[f * NS + r];
    rowmin[e] = 1e9f;
  }

  __shared__ float4 Gl[512];
  __shared__ float Ml[512];

  for (int chunk = 0; chunk < NS; chunk += 512) {
    __syncthreads();
    for (int i = threadIdx.x; i < 512; i += 256) {
      Gl[i] = Gpts[f * NS + chunk + i];
      Ml[i] = Msel[f * NS + chunk + i];
    }
    __syncthreads();
#pragma unroll 2
    for (int jt = 0; jt < 32; ++jt) {
      int col = jt * 16 + laneN;
      float4 gq = Gl[col];
      v2f b;
      b.x = hiK ? gq.z : gq.x;
      b.y = hiK ? 0.0f : gq.y;
      float gn2 = gq.w;
      float cmask = Ml[col];

      v8f c = {};
      c = __builtin_amdgcn_wmma_f32_16x16x4_f32(false, a, false, b, (short)0, c, false, false);

      float colmin = 1e9f;
#pragma unroll
      for (int e = 0; e < 8; ++e) {
        float d2 = pn2[e] + gn2 - 2.0f * c[e];
        float d = fast_sqrtf(fmaxf(d2, 1e-12f));
        rowmin[e] = (cmask > 0.5f) ? fminf(rowmin[e], d) : rowmin[e];
        colmin = fminf(colmin, (rm[e] > 0.5f) ? d : 1e9f);
      }
      // combine both 8-row halves of the tile for this column
      colmin = fminf(colmin, __shfl_xor(colmin, 16, 32));
      atomicMin(&mingp[f * NS + chunk + jt * 16 + laneN], __float_as_uint(colmin));
    }
  }

  // reduce row mins across the 16 lanes of each half (disjoint column residues)
#pragma unroll
  for (int m = 1; m < 16; m <<= 1) {
#pragma unroll
    for (int e = 0; e < 8; ++e)
      rowmin[e] = fminf(rowmin[e], __shfl_xor(rowmin[e], m, 32));
  }
  if (lane == 0) {
#pragma unroll
    for (int e = 0; e < 8; ++e) minpg[f * NS + rowbase + e] = rowmin[e];
  }
  if (lane == 16) {
#pragma unroll
    for (int e = 0; e < 8; ++e) minpg[f * NS + rowbase + 8 + e] = rowmin[e];
  }
}

// ---------------- per-frame Chamfer reduce ----------------
__global__ void chamfer_reduce_kernel(const float* __restrict__ minpg,
                                      const unsigned* __restrict__ mingp,
                                      const float* __restrict__ Msel,
                                      const float* __restrict__ dgt,
                                      float* __restrict__ fcd, float* __restrict__ fok) {
  int f = blockIdx.x, t = threadIdx.x;
  float sw = 0.0f, n1 = 0.0f, n2 = 0.0f;
  for (int i = t; i < NS; i += 256) {
    float w = Msel[f * NS + i];
    sw += w;
    n1 += minpg[f * NS + i] * w;
    n2 += __uint_as_float(mingp[f * NS + i]) * w;
  }
  sw = block_reduce_float(sw);
  n1 = block_reduce_float(n1);
  n2 = block_reduce_float(n2);
  int mc = 0;
  const float* dgf = dgt + (size_t)f * HW;
  for (int i = t; i < HW; i += 256) mc += base_mask(dgf[i]) ? 1 : 0;
  mc = block_reduce_int(mc);
  if (t == 0) {
    float denom = fmaxf(sw, 1.0f);
    float cd = n1 / denom + n2 / denom;
    bool ok = (mc >= 10);
    fcd[f] = ok ? cd : 0.0f;
    fok[f] = ok ? 1.0f : 0.0f;
  }
}

// ---------------- final deterministic combine ----------------
__global__ void final_kernel(const float* __restrict__ acc, const float* __restrict__ part,
                             const float* __restrict__ fcd, const float* __restrict__ fok,
                             float* __restrict__ out) {
  if (blockIdx.x != 0 || threadIdx.x != 0) return;
  float dl = 0.0f, nl = 0.0f, wv = 0.0f;
  for (int i = 0; i < DN_GRID; ++i) {
    dl += part[i * 4 + 0];
    nl += part[i * 4 + 1];
    wv += part[i * 4 + 2];
  }
  float cd = 0.0f, okc = 0.0f;
  for (int f = 0; f < NFR; ++f) { cd += fcd[f]; okc += fok[f]; }
  float denom = fmaxf(wv, 1.0f);
  out[0] = acc[0] + dl / denom + nl / denom + cd / fmaxf(okc, 1.0f);
}

// ---------------- launcher ----------------
extern "C" void kernel_launch(void* const* d_in, const int* in_sizes, int n_in,
                              void* d_out, int out_size, void* d_ws, size_t ws_size,
                              hipStream_t stream) {
  (void)in_sizes; (void)n_in; (void)out_size; (void)ws_size;
  const float* depth_pred  = (const float*)d_in[0];
  const float* points_pred = (const float*)d_in[1];
  const float* depth_gt    = (const float*)d_in[2];
  const float* intrinsics  = (const float*)d_in[3];
  const float* pose_pred   = (const float*)d_in[4];
  const float* pose_gt     = (const float*)d_in[5];
  float* out = (float*)d_out;

  char* p = (char*)d_ws;
  auto carve = [&](size_t bytes) -> char* {
    char* r = p;
    p += (bytes + 255) & ~((size_t)255);
    return r;
  };
  float*    acc   = (float*)carve(16 * sizeof(float));
  float*    scale = (float*)carve(NFR * sizeof(float));
  float*    relgt = (float*)carve(NFR * 12 * sizeof(float));
  int*      sel   = (int*)carve((size_t)NFR * NS * sizeof(int));
  int*      eqbuf = (int*)carve((size_t)NFR * EQCAP * sizeof(int));
  float4*   Ppts  = (float4*)carve((size_t)NFR * NS * sizeof(float4));
  float4*   Gpts  = (float4*)carve((size_t)NFR * NS * sizeof(float4));
  float*    Msel  = (float*)carve((size_t)NFR * NS * sizeof(float));
  float*    minpg = (float*)carve((size_t)NFR * NS * sizeof(float));
  unsigned* mingp = (unsigned*)carve((size_t)NFR * NS * sizeof(unsigned));
  float*    part  = (float*)carve((size_t)DN_GRID * 4 * sizeof(float));
  float*    fcd   = (float*)carve(NFR * sizeof(float));
  float*    fok   = (float*)carve(NFR * sizeof(float));

  init_kernel<<<(NFR * NS + 255) / 256, 256, 0, stream>>>(acc, mingp, sel);
  pose_kernel<<<1, 1, 0, stream>>>(pose_pred, pose_gt, acc, relgt);
  median_scale_kernel<<<NFR, 256, 0, stream>>>(depth_pred, depth_gt, scale);
  depth_normal_kernel<<<DN_GRID, DN_BLK, 0, stream>>>(depth_pred, depth_gt, intrinsics, scale, part);
  select_kernel<<<NFR, 256, 0, stream>>>(depth_gt, sel, eqbuf);
  gather_kernel<<<(NFR * NS + 255) / 256, 256, 0, stream>>>(depth_gt, points_pred, intrinsics,
                                                            scale, relgt, sel, Ppts, Gpts, Msel);
  chamfer_kernel<<<NFR * (NS / 128), 256, 0, stream>>>(Ppts, Gpts, Msel, minpg, mingp);
  chamfer_reduce_kernel<<<NFR, 256, 0, stream>>>(minpg, mingp, Msel, depth_gt, fcd, fok);
  final_kernel<<<1, 1, 0, stream>>>(acc, part, fcd, fok, out);
}